// GCN_84181359001627
// MI455X (gfx1250) — compile-verified
//
#include <hip/hip_runtime.h>
#include <math.h>

// ---------------------------------------------------------------------------
// GCN 2-layer forward for MI455X (gfx1250).
// Dense GEMMs use V_WMMA_F32_16X16X4_F32 (full f32 precision; GEMM cost is
// negligible ~0.5 GFLOP total, so no reason to drop precision).
// Edge aggregation is L2-resident float-atomic scatter (~1.3 GB traffic,
// ~60us at 23.3 TB/s HBM; actually mostly L2 since all buffers fit in 192MB).
// ---------------------------------------------------------------------------

typedef float v2f __attribute__((ext_vector_type(2)));
typedef float v8f __attribute__((ext_vector_type(8)));

#define N_NODES   100000
#define N_EDGES   3200000
#define IN_FEAT   128
#define HIDDEN    16
#define N_CLASSES 40

// ---------------- degree / normalization ----------------

__global__ void gcn_deg_init(float* __restrict__ deg, int n) {
    int i = blockIdx.x * blockDim.x + threadIdx.x;
    if (i < n) deg[i] = 1.0f;   // self-loop contributes 1 to every target degree
}

__global__ void gcn_deg_count(const int* __restrict__ dst, float* __restrict__ deg, int e) {
    int i = blockIdx.x * blockDim.x + threadIdx.x;
    if (i < e) atomicAdd(&deg[dst[i]], 1.0f);
}

__global__ void gcn_dinv(float* __restrict__ dinv, int n) {
    int i = blockIdx.x * blockDim.x + threadIdx.x;
    if (i < n) dinv[i] = rsqrtf(dinv[i]);   // deg >= 1 always (self-loops)
}

// ---------------- GEMM1: h1[N,16] = x[N,128] @ W1[128,16]  (WMMA f32 16x16x4)
// One wave per 16-row tile. 100000 = 6250*16 exactly -> EXEC all-ones, no guards.

__global__ __launch_bounds__(32) void gcn_gemm1(const float* __restrict__ x,
                                                const float* __restrict__ W1,
                                                float* __restrict__ h1) {
    const int lane = threadIdx.x;          // 0..31
    const int m0   = blockIdx.x * 16;
    const int half = lane >> 4;            // 0: K={0,1}, 1: K={2,3}
    const int l    = lane & 15;

    const float* xrow = x + (size_t)(m0 + l) * IN_FEAT;
    v8f acc = {};

    for (int k0 = 0; k0 < IN_FEAT; k0 += 4) {
        const int kb = k0 + half * 2;
        v2f a, b;
        a.x = xrow[kb];
        a.y = xrow[kb + 1];
        b.x = W1[(size_t)(kb    ) * HIDDEN + l];
        b.y = W1[(size_t)(kb + 1) * HIDDEN + l];
        acc = __builtin_amdgcn_wmma_f32_16x16x4_f32(
            false, a, false, b, (short)0, acc, false, false);
    }

    // C/D layout: VGPR r -> row m0 + r + 8*half, col l
#pragma unroll
    for (int r = 0; r < 8; ++r) {
        const int row = m0 + r + 8 * half;
        h1[(size_t)row * HIDDEN + l] = acc[r];
    }
}

// ---------------- GEMM2: h2[N,40] = hrelu[N,16] @ W2[16,40]  (WMMA, 3 N-tiles)

__global__ __launch_bounds__(32) void gcn_gemm2(const float* __restrict__ h,
                                                const float* __restrict__ W2,
                                                float* __restrict__ h2) {
    const int lane = threadIdx.x;
    const int m0   = blockIdx.x * 16;
    const int half = lane >> 4;
    const int l    = lane & 15;

    const float* hrow = h + (size_t)(m0 + l) * HIDDEN;

    for (int nt = 0; nt < 3; ++nt) {
        const int  col   = nt * 16 + l;
        const bool valid = (col < N_CLASSES);
        const int  ccol  = valid ? col : (N_CLASSES - 1);   // clamp: always in-bounds read
        v8f acc = {};
#pragma unroll
        for (int k0 = 0; k0 < HIDDEN; k0 += 4) {
            const int kb = k0 + half * 2;
            v2f a, b;
            a.x = hrow[kb];
            a.y = hrow[kb + 1];
            float bv0 = W2[(size_t)(kb    ) * N_CLASSES + ccol];
            float bv1 = W2[(size_t)(kb + 1) * N_CLASSES + ccol];
            b.x = valid ? bv0 : 0.0f;
            b.y = valid ? bv1 : 0.0f;
            acc = __builtin_amdgcn_wmma_f32_16x16x4_f32(
                false, a, false, b, (short)0, acc, false, false);
        }
        if (valid) {
#pragma unroll
            for (int r = 0; r < 8; ++r) {
                const int row = m0 + r + 8 * half;
                h2[(size_t)row * N_CLASSES + col] = acc[r];
            }
        }
    }
}

// ---------------- aggregation: out = dinv^2 * h  (self-loop folded into init)

template <int F>
__global__ void gcn_agg_init(const float* __restrict__ h,
                             const float* __restrict__ dinv,
                             float* __restrict__ out, int n) {
    long long t = (long long)blockIdx.x * blockDim.x + threadIdx.x;
    if (t >= (long long)n * F) return;
    int i = (int)(t / F);
    float di = dinv[i];
    out[t] = di * di * h[t];
}

// out[dst] += dinv[src]*dinv[dst] * h[src]  — one thread per (edge, feature)

template <int F>
__global__ void gcn_agg_edges(const int* __restrict__ src,
                              const int* __restrict__ dst,
                              const float* __restrict__ h,
                              const float* __restrict__ dinv,
                              float* __restrict__ out, int e) {
    long long t = (long long)blockIdx.x * blockDim.x + threadIdx.x;
    if (t >= (long long)e * F) return;
    int ei = (int)(t / F);
    int f  = (int)(t - (long long)ei * F);
    int s  = src[ei];
    int d  = dst[ei];
    float nrm = dinv[s] * dinv[d];
    atomicAdd(&out[(size_t)d * F + f], nrm * h[(size_t)s * F + f]);
}

// ---------------- bias + relu (layer 1) ----------------

__global__ void gcn_bias_relu(const float* __restrict__ agg,
                              const float* __restrict__ b,
                              float* __restrict__ out, int n) {
    long long t = (long long)blockIdx.x * blockDim.x + threadIdx.x;
    if (t >= (long long)n * HIDDEN) return;
    int f = (int)(t % HIDDEN);
    float v = agg[t] + b[f];
    out[t] = v > 0.0f ? v : 0.0f;
}

// ---------------- bias + log_softmax (layer 2) -> d_out ----------------

__global__ void gcn_logsoftmax(const float* __restrict__ agg,
                               const float* __restrict__ b2,
                               float* __restrict__ out, int n) {
    int i = blockIdx.x * blockDim.x + threadIdx.x;
    if (i >= n) return;
    float z[N_CLASSES];
    float m = -3.0e38f;
#pragma unroll
    for (int c = 0; c < N_CLASSES; ++c) {
        z[c] = agg[(size_t)i * N_CLASSES + c] + b2[c];
        m = fmaxf(m, z[c]);
    }
    float s = 0.0f;
#pragma unroll
    for (int c = 0; c < N_CLASSES; ++c) s += expf(z[c] - m);
    float lse = m + logf(s);
#pragma unroll
    for (int c = 0; c < N_CLASSES; ++c)
        out[(size_t)i * N_CLASSES + c] = z[c] - lse;
}

// ---------------------------------------------------------------------------

extern "C" void kernel_launch(void* const* d_in, const int* in_sizes, int n_in,
                              void* d_out, int out_size, void* d_ws, size_t ws_size,
                              hipStream_t stream) {
    const float* x  = (const float*)d_in[0];
    const int*   ei = (const int*)  d_in[1];   // [2, E] (row 0 = src, row 1 = dst)
    const float* W1 = (const float*)d_in[2];
    const float* b1 = (const float*)d_in[3];
    const float* W2 = (const float*)d_in[4];
    const float* b2 = (const float*)d_in[5];
    float* out = (float*)d_out;

    const int n = N_NODES;
    const int e = N_EDGES;
    const int* src = ei;
    const int* dst = ei + e;

    // workspace carve-up (256B aligned): dinv, h1(/hrelu), agg1, h2, agg2  (~45 MB)
    char*  ws  = (char*)d_ws;
    size_t off = 0;
    auto carve = [&](size_t bytes) -> float* {
        float* p = (float*)(ws + off);
        off = (off + bytes + 255) & ~(size_t)255;
        return p;
    };
    float* dinv = carve((size_t)n * sizeof(float));
    float* h1   = carve((size_t)n * HIDDEN    * sizeof(float));   // reused as hrelu
    float* agg1 = carve((size_t)n * HIDDEN    * sizeof(float));
    float* h2   = carve((size_t)n * N_CLASSES * sizeof(float));
    float* agg2 = carve((size_t)n * N_CLASSES * sizeof(float));

    const int B = 256;
    const int nTiles = n / 16;   // 6250 exactly

    // degrees -> dinv
    gcn_deg_init <<<(n + B - 1) / B, B, 0, stream>>>(dinv, n);
    gcn_deg_count<<<(e + B - 1) / B, B, 0, stream>>>(dst, dinv, e);
    gcn_dinv     <<<(n + B - 1) / B, B, 0, stream>>>(dinv, n);

    // layer 1
    gcn_gemm1<<<nTiles, 32, 0, stream>>>(x, W1, h1);
    {
        long long tot = (long long)n * HIDDEN;
        gcn_agg_init<HIDDEN><<<(unsigned)((tot + B - 1) / B), B, 0, stream>>>(h1, dinv, agg1, n);
        long long etot = (long long)e * HIDDEN;
        gcn_agg_edges<HIDDEN><<<(unsigned)((etot + B - 1) / B), B, 0, stream>>>(src, dst, h1, dinv, agg1, e);
        gcn_bias_relu<<<(unsigned)((tot + B - 1) / B), B, 0, stream>>>(agg1, b1, h1, n);  // hrelu -> h1
    }

    // layer 2
    gcn_gemm2<<<nTiles, 32, 0, stream>>>(h1, W2, h2);
    {
        long long tot = (long long)n * N_CLASSES;
        gcn_agg_init<N_CLASSES><<<(unsigned)((tot + B - 1) / B), B, 0, stream>>>(h2, dinv, agg2, n);
        long long etot = (long long)e * N_CLASSES;
        gcn_agg_edges<N_CLASSES><<<(unsigned)((etot + B - 1) / B), B, 0, stream>>>(src, dst, h2, dinv, agg2, e);
    }

    gcn_logsoftmax<<<(n + B - 1) / B, B, 0, stream>>>(agg2, b2, out, n);

    (void)in_sizes; (void)n_in; (void)out_size; (void)ws_size;
}